// MMD_Loss_39479339385356
// MI455X (gfx1250) — compile-verified
//
#include <hip/hip_runtime.h>
#include <hip/hip_bf16.h>

// MMD loss via Mehler/Hermite factorization:
//   exp(-(u-v)^2/2) = sum_k phi_k(u) phi_k(v),  phi_k(u) = u^k e^{-u^2/2} / sqrt(k!)
// Only the MEAN of each Gram matrix is needed, so everything collapses to
//   loss = (1/D) * sum_{d,k} ( S_s[d,k]/n - S_t[d,k]/m )^2
// with S[d,k] = sum_i phi_k(x_id / sigma).  O(n*D*K) instead of O(n^2*D).

#define DIMS   256
#define KTERMS 64

// compile-time 1/sqrt(k) table (constexpr Newton iteration -> immediates)
struct RskTab { float v[KTERMS]; };
constexpr RskTab make_rsk() {
    RskTab t{};
    t.v[0] = 0.0f;                       // unused (k starts at 1)
    for (int k = 1; k < KTERMS; ++k) {
        double g = (double)k;
        for (int i = 0; i < 64; ++i) g = 0.5 * (g + (double)k / g);
        t.v[k] = (float)(1.0 / g);
    }
    return t;
}
constexpr RskTab RSK = make_rsk();

__device__ __forceinline__ float read_sigma(const void* p) {
    // Python scalar may arrive as int32 or float32; any real float has
    // bits >= 0x00800000, any small positive int is below it.
    unsigned bits = *(const unsigned*)p;
    if (bits < 0x00800000u) return (float)(int)bits;
    return __uint_as_float(bits);
}

// ---------------------------------------------------------------------------
// Kernel A: per-(matrix, dimension) Hermite feature column sums.
// grid = 2*DIMS blocks, 128 threads. Deterministic fixed-order reduction.
// ---------------------------------------------------------------------------
__global__ __launch_bounds__(128) void mmd_feature_sums(
    const float* __restrict__ src, const float* __restrict__ tgt,
    const void* __restrict__ sigp, float* __restrict__ P, int n, int m)
{
    const int  b      = blockIdx.x;              // [0, 2*DIMS)
    const int  d      = b & (DIMS - 1);
    const bool is_tgt = (b >= DIMS);
    const float* __restrict__ X = is_tgt ? tgt : src;
    const int  rows   = is_tgt ? m : n;
    const float inv_sigma = 1.0f / read_sigma(sigp);
    const int  tid    = threadIdx.x;

    float acc[KTERMS];
#pragma unroll
    for (int k = 0; k < KTERMS; ++k) acc[k] = 0.0f;

    for (int i = tid; i < rows; i += 128) {
        const float u = X[i * DIMS + d] * inv_sigma;
        float p = __expf(-0.5f * u * u);         // phi_0
        acc[0] += p;
#pragma unroll
        for (int k = 1; k < KTERMS; ++k) {       // phi_k = phi_{k-1} * u / sqrt(k)
            p *= u * RSK.v[k];
            acc[k] += p;
        }
    }

    // deterministic reduce: shuffle tree within wave32, fixed-order across 4 waves
    __shared__ float part[4][KTERMS];
    const int lane = tid & 31, w = tid >> 5;
#pragma unroll
    for (int k = 0; k < KTERMS; ++k) {
        float r = acc[k];
        for (int off = 16; off > 0; off >>= 1) r += __shfl_xor(r, off, 32);
        if (lane == 0) part[w][k] = r;
    }
    __syncthreads();
    if (tid < KTERMS) {
        float s = (part[0][tid] + part[1][tid]) + (part[2][tid] + part[3][tid]);
        P[b * KTERMS + tid] = s;                 // layout: [mat*DIMS + d][KTERMS]
    }
}

// ---------------------------------------------------------------------------
// Kernel B: loss = (1/D) * || P_s/n - P_t/m ||^2 over 16384 elements,
// computed on the matrix pipe: chunk delta into 16x4 f32 A-operands; SRC1 ==
// SRC0 gives D = A*A^T (A and B share the (lane,vgpr)->K map per the ISA
// layout tables), so sum(diag(D)) == sum(delta^2) for the chunk.
// Single wave32, EXEC all-ones across every wmma.
// ---------------------------------------------------------------------------
typedef __attribute__((ext_vector_type(2))) float v2f;
typedef __attribute__((ext_vector_type(8))) float v8f;

__global__ __launch_bounds__(32) void mmd_finalize_wmma(
    const float* __restrict__ P, float* __restrict__ out, float invn, float invm)
{
    const int lane = threadIdx.x;
    const int mrow = lane & 15;                  // A-row (M) held by this lane
    const int h    = lane >> 4;                  // half: K base = 2h
    const float* __restrict__ Ps = P;
    const float* __restrict__ Pt = P + DIMS * KTERMS;

    v8f c = {};
    const int L = DIMS * KTERMS;                 // 16384
    for (int base = 0; base < L; base += 64) {
        // A[mrow][k] = delta[base + 4*mrow + k]; this lane holds k = 2h, 2h+1
        const int j0 = base + mrow * 4 + 2 * h;
        v2f a;
        a.x = Ps[j0]     * invn - Pt[j0]     * invm;
        a.y = Ps[j0 + 1] * invn - Pt[j0 + 1] * invm;
        c = __builtin_amdgcn_wmma_f32_16x16x4_f32(
                false, a, false, a, (short)0, c, false, false);
    }

    // diag(D): (VGPR j, lane j) holds D[j][j]; (VGPR j, lane j+24) holds D[j+8][j+8]
    float diag = 0.0f;
#pragma unroll
    for (int j = 0; j < 8; ++j)
        diag += ((lane == j) || (lane == j + 24)) ? c[j] : 0.0f;
    for (int off = 16; off > 0; off >>= 1) diag += __shfl_xor(diag, off, 32);

    if (lane == 0) out[0] = diag * (1.0f / (float)DIMS);
}

extern "C" void kernel_launch(void* const* d_in, const int* in_sizes, int n_in,
                              void* d_out, int out_size, void* d_ws, size_t ws_size,
                              hipStream_t stream) {
    (void)n_in; (void)out_size; (void)ws_size;
    const float* src = (const float*)d_in[0];
    const float* tgt = (const float*)d_in[1];
    const void*  sig = d_in[2];
    const int n = in_sizes[0] / DIMS;            // 1024
    const int m = in_sizes[1] / DIMS;            // 1024

    float* P = (float*)d_ws;                     // 2 * DIMS * KTERMS floats = 128 KB

    mmd_feature_sums<<<2 * DIMS, 128, 0, stream>>>(src, tgt, sig, P, n, m);
    mmd_finalize_wmma<<<1, 32, 0, stream>>>(P, (float*)d_out,
                                            1.0f / (float)n, 1.0f / (float)m);
}